// MultiHeadSelfAttention_42460046688743
// MI455X (gfx1250) — compile-verified
//
#include <hip/hip_runtime.h>
#include <hip/hip_bf16.h>

typedef __attribute__((ext_vector_type(2))) float v2f;
typedef __attribute__((ext_vector_type(8))) float v8f;

#define WMMA_F32(a, b, c) \
  __builtin_amdgcn_wmma_f32_16x16x4_f32(false, (a), false, (b), (short)0, (c), false, false)

#define BB 8
#define SS 1024
#define DD 1024
#define HH 16
#define DK 64

__device__ __forceinline__ float grp16_max(float v) {
  v = fmaxf(v, __shfl_xor(v, 1, 32));
  v = fmaxf(v, __shfl_xor(v, 2, 32));
  v = fmaxf(v, __shfl_xor(v, 4, 32));
  v = fmaxf(v, __shfl_xor(v, 8, 32));
  return v;
}
__device__ __forceinline__ float grp16_sum(float v) {
  v += __shfl_xor(v, 1, 32);
  v += __shfl_xor(v, 2, 32);
  v += __shfl_xor(v, 4, 32);
  v += __shfl_xor(v, 8, 32);
  return v;
}

// ---------------------------------------------------------------------------
// Generic GEMM: out = A[8192 x 1024] * W[1024 x 1024] + bias[1024]
// mode 0: plain [8192 x 1024] store  (Q, V, O projections / final output)
// mode 1: store transposed-per-head:  Kt[b][h][d][s]  (for coalesced QK^T)
// grid = (512 row-tiles, 8 col-supertiles), block = 256 (8 waves).
// Each wave owns one 16x16 output tile; A tile (16x64) staged in LDS.
// ---------------------------------------------------------------------------
__global__ __launch_bounds__(256) void gemm_bias_kernel(
    const float* __restrict__ A, const float* __restrict__ W,
    const float* __restrict__ bias, float* __restrict__ out, int mode) {
  __shared__ __align__(16) float lx[16 * 68];   // A tile, padded stride 68
  __shared__ float lt[8 * 16 * 17];             // per-wave transpose staging

  const int tid  = threadIdx.x;
  const int wave = tid >> 5;
  const int lane = tid & 31;
  const int l16  = lane & 15;
  const int hi   = lane >> 4;          // 0: lanes 0-15, 1: lanes 16-31
  const int rowBase = blockIdx.x * 16;
  const int colBase = blockIdx.y * 128 + wave * 16;

  v8f acc = {};

  for (int kk = 0; kk < DD; kk += 64) {
    // cooperative coalesced load of A tile: 16 rows x 64 cols
    {
      int i = tid * 4;                 // 256 threads * 4 floats = 1024
      int r = i >> 6, c = i & 63;
      const float4 src = *(const float4*)(A + (size_t)(rowBase + r) * DD + kk + c);
      *(float4*)&lx[r * 68 + c] = src;
    }
    __syncthreads();
#pragma unroll
    for (int c = 0; c < 16; ++c) {
      int k = c * 4 + hi * 2;
      v2f a = *(const v2f*)&lx[l16 * 68 + k];
      v2f b;
      b.x = W[(size_t)(kk + k) * DD + colBase + l16];
      b.y = W[(size_t)(kk + k + 1) * DD + colBase + l16];
      acc = WMMA_F32(a, b, acc);
    }
    __syncthreads();
  }

  const float bv = bias[colBase + l16];
#pragma unroll
  for (int g = 0; g < 8; ++g) acc[g] += bv;

  if (mode == 0) {
#pragma unroll
    for (int g = 0; g < 8; ++g) {
      int M = g + hi * 8;
      out[(size_t)(rowBase + M) * DD + colBase + l16] = acc[g];
    }
  } else {
    // K^T store: element (row=s_global, col) -> Kt[b][h][d][s]
    float* T = &lt[wave * 16 * 17];
#pragma unroll
    for (int g = 0; g < 8; ++g) T[(g + hi * 8) * 17 + l16] = acc[g];
    const int b = rowBase / SS, sBase = rowBase % SS;
    const int h = colBase / DK, dBase = colBase % DK;
    float* kt = out + (((size_t)(b * HH + h) * DK + dBase) * SS) + sBase;
#pragma unroll
    for (int dd = 0; dd < 8; ++dd) {
      int dl = dd + hi * 8;
      kt[(size_t)dl * SS + l16] = T[l16 * 17 + dl];  // lanes 0-15 contiguous
    }
  }
}

// ---------------------------------------------------------------------------
// Fused attention (flash-style streaming, exact post-softmax mask + renorm):
//   attn = e * m / (sum(e*m) + eps * sum(e)),  e = exp(s - rowmax)
// One wave = one (b,h, 16-query tile). 8 waves / block, 1024 blocks.
// ---------------------------------------------------------------------------
__global__ __launch_bounds__(256) void attn_fused_kernel(
    const float* __restrict__ Qb,   // [B,S,D] plain
    const float* __restrict__ Kt,   // [B*H, DK, S]
    const float* __restrict__ Vb,   // [B,S,D] plain
    const int* __restrict__ mask,   // [B,1,S,S]
    float* __restrict__ ctx) {      // [B,S,D]
  __shared__ float lp[8 * 16 * 18]; // per-wave P tile staging (16x16, pad 18)

  const int tid  = threadIdx.x;
  const int wave = tid >> 5;
  const int lane = tid & 31;
  const int l16  = lane & 15;
  const int hi   = lane >> 4;

  const int task = blockIdx.x * 8 + wave;   // 8192 tasks
  const int qt = task & 63;
  const int bh = task >> 6;
  const int b = bh >> 4, h = bh & 15;
  const int qbase = qt * 16;

  // Q tile as A-fragments (16 chunks of K=4), loaded once, reused 64x
  v2f qf[16];
  const float* qrow = Qb + ((size_t)(b * SS) + qbase + l16) * DD + h * DK;
#pragma unroll
  for (int c = 0; c < 16; ++c) qf[c] = *(const v2f*)(qrow + c * 4 + hi * 2);

  float m[8], Z[8], Zm[8];
  v8f acc[4] = {{}, {}, {}, {}};
#pragma unroll
  for (int g = 0; g < 8; ++g) { m[g] = -1e30f; Z[g] = 0.f; Zm[g] = 0.f; }

  const float* ktB = Kt + (size_t)bh * DK * SS;
  const float* vB  = Vb + (size_t)(b * SS) * DD + h * DK;
  const int*   mB  = mask + (size_t)b * SS * SS + (size_t)qbase * SS;
  float* P = &lp[wave * 16 * 18];

  for (int j = 0; j < 64; ++j) {
    const int kb = j * 16;
    // S_tile = Q * K^T  (coalesced loads from transposed K)
    v8f s = {};
#pragma unroll
    for (int c = 0; c < 16; ++c) {
      int r = c * 4 + hi * 2;
      v2f kf;
      kf.x = ktB[(size_t)r * SS + kb + l16];
      kf.y = ktB[(size_t)(r + 1) * SS + kb + l16];
      s = WMMA_F32(qf[c], kf, s);
    }
    // online softmax bookkeeping (per 16-lane row group), masked sums
    float resc[8];
#pragma unroll
    for (int g = 0; g < 8; ++g) {
      float sv = s[g] * 0.125f;                  // 1/sqrt(64)
      float mn = fmaxf(m[g], grp16_max(sv));
      float rs = __expf(m[g] - mn);
      float p  = __expf(sv - mn);
      int   mk = mB[(size_t)(g + hi * 8) * SS + kb + l16];
      float pm = mk ? p : 0.f;
      Z[g]  = Z[g]  * rs + grp16_sum(p);
      Zm[g] = Zm[g] * rs + grp16_sum(pm);
      m[g] = mn;
      resc[g] = rs;
      P[(g + hi * 8) * 18 + l16] = pm;           // stage P for A-layout reload
    }
#pragma unroll
    for (int nt = 0; nt < 4; ++nt) {
      v8f a = acc[nt];
#pragma unroll
      for (int g = 0; g < 8; ++g) a[g] *= resc[g];
      acc[nt] = a;
    }
    // reload P in A-fragment layout (same-wave LDS; hw waits inserted)
    v2f pf[4];
#pragma unroll
    for (int c = 0; c < 4; ++c) pf[c] = *(const v2f*)&P[l16 * 18 + c * 4 + hi * 2];
    // acc += P * V
#pragma unroll
    for (int nt = 0; nt < 4; ++nt) {
      v8f a = acc[nt];
#pragma unroll
      for (int c = 0; c < 4; ++c) {
        int r = kb + c * 4 + hi * 2;
        v2f vf;
        vf.x = vB[(size_t)r * DD + nt * 16 + l16];
        vf.y = vB[(size_t)(r + 1) * DD + nt * 16 + l16];
        a = WMMA_F32(pf[c], vf, a);
      }
      acc[nt] = a;
    }
  }

  // epilogue: ctx = acc / (Zm + eps*Z)   (all-zero-mask rows -> 0, matches ref)
#pragma unroll
  for (int g = 0; g < 8; ++g) {
    float inv = 1.0f / (Zm[g] + 1e-8f * Z[g]);
    int M = g + hi * 8;
    float* orow = ctx + ((size_t)(b * SS) + qbase + M) * DD + h * DK;
#pragma unroll
    for (int nt = 0; nt < 4; ++nt) orow[nt * 16 + l16] = acc[nt][g] * inv;
  }
}

// ---------------------------------------------------------------------------
extern "C" void kernel_launch(void* const* d_in, const int* in_sizes, int n_in,
                              void* d_out, int out_size, void* d_ws, size_t ws_size,
                              hipStream_t stream) {
  const float* X    = (const float*)d_in[0];
  const int*   mask = (const int*)d_in[1];
  const float* Wq   = (const float*)d_in[2];
  const float* bq   = (const float*)d_in[3];
  const float* Wk   = (const float*)d_in[4];
  const float* bk   = (const float*)d_in[5];
  const float* Wv   = (const float*)d_in[6];
  const float* bv   = (const float*)d_in[7];
  const float* Wo   = (const float*)d_in[8];
  const float* bo   = (const float*)d_in[9];
  float* out = (float*)d_out;

  const size_t MAT = (size_t)BB * SS * DD;   // 8.39M floats per buffer
  float* Qp = (float*)d_ws;
  float* Kp = Qp + MAT;   // transposed per-head layout
  float* Vp = Kp + MAT;
  float* Cx = Vp + MAT;

  dim3 blk(256);
  dim3 grid(512, 8);
  gemm_bias_kernel<<<grid, blk, 0, stream>>>(X, Wq, bq, Qp, 0);
  gemm_bias_kernel<<<grid, blk, 0, stream>>>(X, Wk, bk, Kp, 1);
  gemm_bias_kernel<<<grid, blk, 0, stream>>>(X, Wv, bv, Vp, 0);
  attn_fused_kernel<<<dim3(1024), blk, 0, stream>>>(Qp, Kp, Vp, mask, Cx);
  gemm_bias_kernel<<<grid, blk, 0, stream>>>(Cx, Wo, bo, out, 0);
}